// S_classify_12137577578914
// MI455X (gfx1250) — compile-verified
//
#include <hip/hip_runtime.h>
#include <hip/hip_bf16.h>

// ---------------------------------------------------------------------------
// 2-layer GCN (PyG GCNConv semantics) + Gumbel softmax, for gfx1250 (CDNA5).
// Dense transform x@W1: V_WMMA_F32_16X16X4_F32 (native fp32 matrix path,
// wave32), W1 staged to LDS via TENSOR_LOAD_TO_LDS (TDM) with a
// float4-copy backstop. Edge aggregation: global f32 atomics (L2-side RMW).
// ---------------------------------------------------------------------------

typedef float v2f __attribute__((ext_vector_type(2)));
typedef float v8f __attribute__((ext_vector_type(8)));
typedef unsigned int u32x4 __attribute__((ext_vector_type(4)));
typedef int i32x4 __attribute__((ext_vector_type(4)));
typedef int i32x8 __attribute__((ext_vector_type(8)));

#define D_FEAT 128
#define GUMBEL_INV_TEMP 2.0f
#define EPS 1e-9f

#if defined(__gfx1250__) && __has_builtin(__builtin_amdgcn_tensor_load_to_lds)
#define USE_TDM 1
#else
#define USE_TDM 0
#endif

#if __has_include(<hip/amd_detail/amd_gfx1250_TDM.h>)
#define TDM_6ARG 1   // therock-10.0 headers -> 6-arg builtin
#else
#define TDM_6ARG 0   // ROCm 7.2 -> 5-arg builtin
#endif

// ---------------- degree / normalization ----------------

__global__ void k_deg_init(float* __restrict__ deg, int n) {
    int i = blockIdx.x * blockDim.x + threadIdx.x;
    if (i < n) deg[i] = 1.0f;   // self-loop contribution
}

__global__ void k_deg_edges(const int* __restrict__ dst, float* __restrict__ deg, int e) {
    int i = blockIdx.x * blockDim.x + threadIdx.x;
    if (i < e) atomicAdd(&deg[dst[i]], 1.0f);
}

__global__ void k_dinv(float* __restrict__ deg, int n) {
    int i = blockIdx.x * blockDim.x + threadIdx.x;
    if (i < n) deg[i] = rsqrtf(deg[i]);   // deg >= 1 always (self-loops)
}

// ---------------- WMMA GEMM: H = X @ W  (K = N = 128) ----------------
// One wave -> 32 rows x 128 cols (two 16-row M-tiles share each B fragment,
// halving LDS traffic per v_wmma). W staged in LDS (128x128 f32 = 64 KB).
// __launch_bounds__(256,1): 8 wave32 / WGP -> ~256 VGPR budget per wave so
// all 16 v8f accumulators (128 VGPRs) stay resident (no scratch spills).
// A fragment (16x4 f32): lanes 0-15 carry M=lane (K pair 0,1),
//                        lanes 16-31 carry M=lane-16 (K pair 2,3).
// C/D (16x16 f32, 8 VGPRs): VGPR t -> row t + 8*kh, col = lane&15.

__global__ void __launch_bounds__(256, 1)
k_gemm_wmma_f32(const float* __restrict__ X,
                const float* __restrict__ W,
                float* __restrict__ H,
                int nRows, int nStripes) {
    extern __shared__ float w_lds[];     // 128*128 floats = 64 KB dynamic LDS

#if USE_TDM
    // ---- TDM stage: TENSOR_LOAD_TO_LDS of the full 128x128 f32 tile ----
    if (threadIdx.x < 32) {              // one full wave issues (EXEC all-ones)
        const unsigned lds_off = (unsigned)(size_t)(void*)w_lds;  // LDS byte offset
        const unsigned long long ga = (unsigned long long)(size_t)W;
        // D# group 0: count=1 | lds_addr | global_addr(57b) | type=2
        u32x4 g0;
        g0[0] = 1u;
        g0[1] = lds_off;
        g0[2] = (unsigned)(ga & 0xffffffffu);
        g0[3] = (unsigned)((ga >> 32) & 0x01ffffffu) | (2u << 30);
        // D# group 1: data_size=4B, tensor 128x128, tile 128x128, stride0=128
        i32x8 g1;
        g1[0] = (int)(2u << 16);          // data_size = 4 bytes
        g1[1] = (int)(128u << 16);        // tensor_dim0[15:0]
        g1[2] = (int)(128u << 16);        // tensor_dim1[15:0]
        g1[3] = (int)(128u << 16);        // tile_dim0
        g1[4] = 128;                      // tile_dim1 (tile_dim2 = 0)
        g1[5] = 128;                      // tensor_dim0_stride[31:0]
        g1[6] = 0;
        g1[7] = 0;
        i32x4 gz = {0, 0, 0, 0};
#if TDM_6ARG
        i32x8 gz8 = {0, 0, 0, 0, 0, 0, 0, 0};
        __builtin_amdgcn_tensor_load_to_lds(g0, g1, gz, gz, gz8, 0);
#else
        __builtin_amdgcn_tensor_load_to_lds(g0, g1, gz, gz, 0);
#endif
        __builtin_amdgcn_s_wait_tensorcnt(0);   // s_wait_tensorcnt 0
    }
    __syncthreads();
#endif
    // Staging (backstop: rewrites same bytes, fully coalesced)
    {
        const float4* Wv = (const float4*)W;
        float4* Lv = (float4*)w_lds;
        for (int t = threadIdx.x; t < (D_FEAT * D_FEAT) / 4; t += blockDim.x)
            Lv[t] = Wv[t];
    }
    __syncthreads();

    const int waveInBlock = threadIdx.x >> 5;
    const int wave = blockIdx.x * (blockDim.x >> 5) + waveInBlock;
    if (wave >= nStripes) return;

    const int lane = threadIdx.x & 31;
    const int r  = lane & 15;      // row within tile (A) / col within tile (B,D)
    const int kh = lane >> 4;      // selects K pair {0,1} vs {2,3}

    const int row0 = wave * 32;    // two 16-row M-tiles per wave
    int arow0 = row0 + r;
    int arow1 = row0 + 16 + r;
    if (arow0 >= nRows) arow0 = nRows - 1;   // clamp reads; stores are guarded
    if (arow1 >= nRows) arow1 = nRows - 1;

    v8f acc0[8] = {};
    v8f acc1[8] = {};

    for (int kb = 0; kb < D_FEAT; kb += 4) {
        // A fragments: 2 consecutive f32 along K (8-byte aligned)
        v2f a0 = *(const v2f*)(X + (size_t)arow0 * D_FEAT + kb + 2 * kh);
        v2f a1 = *(const v2f*)(X + (size_t)arow1 * D_FEAT + kb + 2 * kh);

#pragma unroll
        for (int nt = 0; nt < 8; ++nt) {
            const int col = nt * 16 + r;
            v2f b;
            b[0] = w_lds[(kb + 2 * kh + 0) * D_FEAT + col];
            b[1] = w_lds[(kb + 2 * kh + 1) * D_FEAT + col];
            acc0[nt] = __builtin_amdgcn_wmma_f32_16x16x4_f32(
                false, a0, false, b, (short)0, acc0[nt], false, false);
            acc1[nt] = __builtin_amdgcn_wmma_f32_16x16x4_f32(
                false, a1, false, b, (short)0, acc1[nt], false, false);
        }
    }

#pragma unroll
    for (int nt = 0; nt < 8; ++nt) {
#pragma unroll
        for (int t = 0; t < 8; ++t) {
            const int rowA = row0 + t + 8 * kh;
            const int rowB = row0 + 16 + t + 8 * kh;
            if (rowA < nRows)
                H[(size_t)rowA * D_FEAT + nt * 16 + r] = acc0[nt][t];
            if (rowB < nRows)
                H[(size_t)rowB * D_FEAT + nt * 16 + r] = acc1[nt][t];
        }
    }
}

// ---------------- layer-1 aggregation ----------------
// agg[i][d] = h[i][d] * dinv[i]^2 + b[d]   (self-loop + bias, also zero-init)

__global__ void k_self_bias_128(const float* __restrict__ h,
                                const float* __restrict__ dinv,
                                const float* __restrict__ b,
                                float* __restrict__ agg, int n) {
    int i = blockIdx.x * blockDim.x + threadIdx.x;
    if (i >= n * D_FEAT) return;
    const int node = i >> 7;
    const int d = i & (D_FEAT - 1);
    const float w = dinv[node] * dinv[node];
    agg[i] = h[i] * w + b[d];
}

// one wave per edge; each lane handles 4 consecutive features
__global__ void k_edge_agg_128(const int* __restrict__ src,
                               const int* __restrict__ dst,
                               const float* __restrict__ dinv,
                               const float* __restrict__ h,
                               float* __restrict__ agg, int e) {
    const int tid = blockIdx.x * blockDim.x + threadIdx.x;
    const int edge = tid >> 5;
    if (edge >= e) return;
    const int lane = tid & 31;
    const int s = src[edge];
    const int d = dst[edge];
    const float w = dinv[s] * dinv[d];
    const float4 v = *(const float4*)(h + (size_t)s * D_FEAT + lane * 4);
    float* p = agg + (size_t)d * D_FEAT + lane * 4;
    atomicAdd(p + 0, v.x * w);
    atomicAdd(p + 1, v.y * w);
    atomicAdd(p + 2, v.z * w);
    atomicAdd(p + 3, v.w * w);
}

// ---------------- layer 2: 128 -> 2 ----------------

__global__ void k_gemm2(const float* __restrict__ A,   // [n,128]
                        const float* __restrict__ W2,  // [128,2] row-major
                        float* __restrict__ out,       // [n,2]
                        int n) {
    int i = blockIdx.x * blockDim.x + threadIdx.x;
    if (i >= n) return;
    float acc0 = 0.0f, acc1 = 0.0f;
    const float4* row = (const float4*)(A + (size_t)i * D_FEAT);
#pragma unroll 4
    for (int k4 = 0; k4 < D_FEAT / 4; ++k4) {
        const float4 v = row[k4];
        const int k = k4 * 4;
        acc0 += v.x * W2[(k + 0) * 2 + 0]; acc1 += v.x * W2[(k + 0) * 2 + 1];
        acc0 += v.y * W2[(k + 1) * 2 + 0]; acc1 += v.y * W2[(k + 1) * 2 + 1];
        acc0 += v.z * W2[(k + 2) * 2 + 0]; acc1 += v.z * W2[(k + 2) * 2 + 1];
        acc0 += v.w * W2[(k + 3) * 2 + 0]; acc1 += v.w * W2[(k + 3) * 2 + 1];
    }
    out[2 * i + 0] = acc0;
    out[2 * i + 1] = acc1;
}

__global__ void k_self_bias_2(const float* __restrict__ lpre,
                              const float* __restrict__ dinv,
                              const float* __restrict__ b,
                              float* __restrict__ agg, int n) {
    int i = blockIdx.x * blockDim.x + threadIdx.x;
    if (i >= n) return;
    const float w = dinv[i] * dinv[i];
    agg[2 * i + 0] = lpre[2 * i + 0] * w + b[0];
    agg[2 * i + 1] = lpre[2 * i + 1] * w + b[1];
}

__global__ void k_edge_agg_2(const int* __restrict__ src,
                             const int* __restrict__ dst,
                             const float* __restrict__ dinv,
                             const float* __restrict__ lpre,
                             float* __restrict__ agg, int e) {
    int i = blockIdx.x * blockDim.x + threadIdx.x;
    if (i >= e) return;
    const int s = src[i];
    const int d = dst[i];
    const float w = dinv[s] * dinv[d];
    atomicAdd(&agg[2 * d + 0], lpre[2 * s + 0] * w);
    atomicAdd(&agg[2 * d + 1], lpre[2 * s + 1] * w);
}

// ---------------- gumbel softmax ----------------

__global__ void k_gumbel_softmax(const float* __restrict__ logits,
                                 const float* __restrict__ noise,
                                 float* __restrict__ out, int n) {
    int i = blockIdx.x * blockDim.x + threadIdx.x;
    if (i >= n) return;
    const float g0 = -logf(-logf(noise[2 * i + 0] + EPS) + EPS);
    const float g1 = -logf(-logf(noise[2 * i + 1] + EPS) + EPS);
    const float z0 = (logits[2 * i + 0] + g0) * GUMBEL_INV_TEMP;
    const float z1 = (logits[2 * i + 1] + g1) * GUMBEL_INV_TEMP;
    const float m = fmaxf(z0, z1);
    const float e0 = expf(z0 - m);
    const float e1 = expf(z1 - m);
    const float inv = 1.0f / (e0 + e1);
    out[2 * i + 0] = e0 * inv;
    out[2 * i + 1] = e1 * inv;
}

// ---------------------------------------------------------------------------

extern "C" void kernel_launch(void* const* d_in, const int* in_sizes, int n_in,
                              void* d_out, int out_size, void* d_ws, size_t ws_size,
                              hipStream_t stream) {
    const float* x     = (const float*)d_in[0];
    const int*   ei    = (const int*)d_in[1];
    const float* W1    = (const float*)d_in[2];
    const float* b1    = (const float*)d_in[3];
    const float* W2    = (const float*)d_in[4];
    const float* b2    = (const float*)d_in[5];
    const float* noise = (const float*)d_in[6];
    float* out = (float*)d_out;

    const int N = in_sizes[0] / D_FEAT;
    const int E = in_sizes[1] / 2;
    const int* src = ei;        // edge_index[0]
    const int* dst = ei + E;    // edge_index[1]

    // workspace layout (floats)
    float* ws = (float*)d_ws;
    size_t o = 0;
    float* dinv = ws + o; o += (size_t)((N + 255) / 256) * 256;
    float* h    = ws + o; o += (size_t)N * D_FEAT;   // x @ W1
    float* agg1 = ws + o; o += (size_t)N * D_FEAT;   // layer-1 output
    float* lpre = ws + o; o += (size_t)2 * N;        // agg1 @ W2
    float* agg2 = ws + o; o += (size_t)2 * N;        // layer-2 logits
    (void)ws_size; (void)n_in; (void)out_size;

    const int B = 256;

    // ---- normalization: deg -> dinv ----
    k_deg_init <<<(N + B - 1) / B, B, 0, stream>>>(dinv, N);
    k_deg_edges<<<(E + B - 1) / B, B, 0, stream>>>(dst, dinv, E);
    k_dinv     <<<(N + B - 1) / B, B, 0, stream>>>(dinv, N);

    // ---- layer 1: dense transform (WMMA) + normalized aggregation ----
    const int nStripes = (N + 31) / 32;            // 32 rows per wave
    const int wavesPerBlock = B / 32;
    const int gemmBlocks = (nStripes + wavesPerBlock - 1) / wavesPerBlock;
    const size_t ldsBytes = (size_t)D_FEAT * D_FEAT * sizeof(float);  // 64 KB
    k_gemm_wmma_f32<<<gemmBlocks, B, ldsBytes, stream>>>(x, W1, h, N, nStripes);

    k_self_bias_128<<<((size_t)N * D_FEAT + B - 1) / B, B, 0, stream>>>(h, dinv, b1, agg1, N);
    k_edge_agg_128 <<<((size_t)E * 32 + B - 1) / B, B, 0, stream>>>(src, dst, dinv, h, agg1, E);

    // ---- layer 2: 128->2 transform + aggregation ----
    k_gemm2      <<<(N + B - 1) / B, B, 0, stream>>>(agg1, W2, lpre, N);
    k_self_bias_2<<<(N + B - 1) / B, B, 0, stream>>>(lpre, dinv, b2, agg2, N);
    k_edge_agg_2 <<<(E + B - 1) / B, B, 0, stream>>>(src, dst, dinv, lpre, agg2, E);

    // ---- gumbel softmax ----
    k_gumbel_softmax<<<(N + B - 1) / B, B, 0, stream>>>(agg2, noise, out, N);
}